// LatencyModel_18846316494852
// MI455X (gfx1250) — compile-verified
//
#include <hip/hip_runtime.h>

#define N_NODES 8192
#define N_EDGES 262144
#define EMBED   128
#define HID1    64
#define HID2    32
#define EDGE_DIM 16
#define GEPS    1e-9f
#define SLOPE   0.01f

typedef __attribute__((ext_vector_type(2))) float v2f;
typedef __attribute__((ext_vector_type(8))) float v8f;

// ---------------- init / gather ----------------

__global__ void k_zero(float* __restrict__ p, int n) {
    int i = blockIdx.x * blockDim.x + threadIdx.x;
    if (i < n) p[i] = 0.0f;
}

__global__ void k_gather(const int* __restrict__ x, const float* __restrict__ emb,
                         float* __restrict__ h0) {
    int i = blockIdx.x * blockDim.x + threadIdx.x;   // N_NODES*EMBED threads
    int n = i >> 7;
    int j = i & (EMBED - 1);
    h0[i] = emb[x[n] * EMBED + j];
}

// ---------------- edge pass 1: m = relu(h0[src] + (ea@E1w + E1b)); agg1[dst] += m ----------------
// one edge per 128-thread block -> edge_attr loads are block-uniform (scalarizable)

__global__ void k_edge1(const int* __restrict__ ei, const float* __restrict__ ea,
                        const float* __restrict__ E1w, const float* __restrict__ E1b,
                        const float* __restrict__ h0, float* __restrict__ agg1) {
    int e = blockIdx.x;
    int j = threadIdx.x;                     // 0..127
    int src = ei[e];
    int dst = ei[N_EDGES + e];
    float acc = E1b[j];
#pragma unroll
    for (int k = 0; k < EDGE_DIM; ++k)
        acc = fmaf(ea[e * EDGE_DIM + k], E1w[k * EMBED + j], acc);
    float m = h0[src * EMBED + j] + acc;
    m = fmaxf(m, 0.0f);
    unsafeAtomicAdd(&agg1[dst * EMBED + j], m);
}

// ---------------- node update 1: h1 = leaky_relu(((1+eps)*h0 + agg1) @ W1 + b1) ----------------
// one wave per 16x16 tile; fp32 WMMA 16x16x4, K=128 -> 32 wmma per wave

__global__ void k_update1(const float* __restrict__ h0, const float* __restrict__ agg1,
                          const float* __restrict__ W1, const float* __restrict__ b1,
                          float* __restrict__ h1) {
    int wave = (blockIdx.x * blockDim.x + threadIdx.x) >> 5;
    int lane = threadIdx.x & 31;
    int mt = wave >> 2;            // 0..511
    int nt = wave & 3;             // 0..3
    int hi  = lane >> 4;           // 0: K pair {k,k+1}, 1: {k+2,k+3}
    int lid = lane & 15;
    int rowA = mt * 16 + lid;
    int colB = nt * 16 + lid;
    v8f acc = {};
#pragma unroll
    for (int k = 0; k < EMBED; k += 4) {
        int kk = k + hi * 2;
        v2f hv = *(const v2f*)(h0   + rowA * EMBED + kk);
        v2f gv = *(const v2f*)(agg1 + rowA * EMBED + kk);
        v2f a, b;
        a.x = fmaf(1.0f + GEPS, hv.x, gv.x);
        a.y = fmaf(1.0f + GEPS, hv.y, gv.y);
        b.x = W1[kk * HID1 + colB];
        b.y = W1[(kk + 1) * HID1 + colB];
        acc = __builtin_amdgcn_wmma_f32_16x16x4_f32(false, a, false, b, (short)0, acc, false, false);
    }
    float bias = b1[colB];
#pragma unroll
    for (int r = 0; r < 8; ++r) {
        int row = mt * 16 + hi * 8 + r;
        float v = acc[r] + bias;
        v = (v >= 0.0f) ? v : SLOPE * v;
        h1[row * HID1 + colB] = v;
    }
}

// ---------------- edge pass 2 ----------------

__global__ void k_edge2(const int* __restrict__ ei, const float* __restrict__ ea,
                        const float* __restrict__ E2w, const float* __restrict__ E2b,
                        const float* __restrict__ h1, float* __restrict__ agg2) {
    int e = blockIdx.x;
    int j = threadIdx.x;                     // 0..63
    int src = ei[e];
    int dst = ei[N_EDGES + e];
    float acc = E2b[j];
#pragma unroll
    for (int k = 0; k < EDGE_DIM; ++k)
        acc = fmaf(ea[e * EDGE_DIM + k], E2w[k * HID1 + j], acc);
    float m = h1[src * HID1 + j] + acc;
    m = fmaxf(m, 0.0f);
    unsafeAtomicAdd(&agg2[dst * HID1 + j], m);
}

// ---------------- node update 2: h2 = ((1+eps)*h1 + agg2) @ W2 + b2 ----------------

__global__ void k_update2(const float* __restrict__ h1, const float* __restrict__ agg2,
                          const float* __restrict__ W2, const float* __restrict__ b2,
                          float* __restrict__ h2) {
    int wave = (blockIdx.x * blockDim.x + threadIdx.x) >> 5;
    int lane = threadIdx.x & 31;
    int mt = wave >> 1;            // 0..511
    int nt = wave & 1;             // 0..1
    int hi  = lane >> 4;
    int lid = lane & 15;
    int rowA = mt * 16 + lid;
    int colB = nt * 16 + lid;
    v8f acc = {};
#pragma unroll
    for (int k = 0; k < HID1; k += 4) {
        int kk = k + hi * 2;
        v2f hv = *(const v2f*)(h1   + rowA * HID1 + kk);
        v2f gv = *(const v2f*)(agg2 + rowA * HID1 + kk);
        v2f a, b;
        a.x = fmaf(1.0f + GEPS, hv.x, gv.x);
        a.y = fmaf(1.0f + GEPS, hv.y, gv.y);
        b.x = W2[kk * HID2 + colB];
        b.y = W2[(kk + 1) * HID2 + colB];
        acc = __builtin_amdgcn_wmma_f32_16x16x4_f32(false, a, false, b, (short)0, acc, false, false);
    }
    float bias = b2[colB];
#pragma unroll
    for (int r = 0; r < 8; ++r) {
        int row = mt * 16 + hi * 8 + r;
        h2[row * HID2 + colB] = acc[r] + bias;
    }
}

// ---------------- out = h2 @ h2^T : one wave per 16x16 output tile ----------------

__global__ void k_outer(const float* __restrict__ h2, float* __restrict__ out) {
    int wave = (blockIdx.x * blockDim.x + threadIdx.x) >> 5;
    int lane = threadIdx.x & 31;
    int mt = wave >> 9;            // 0..511
    int nt = wave & 511;           // 0..511
    int hi  = lane >> 4;
    int lid = lane & 15;
    const float* arow = h2 + (size_t)(mt * 16 + lid) * HID2;
    const float* brow = h2 + (size_t)(nt * 16 + lid) * HID2;   // B[k][n] = h2[n][k]
    v8f acc = {};
#pragma unroll
    for (int k = 0; k < HID2; k += 4) {
        int kk = k + hi * 2;
        v2f a = *(const v2f*)(arow + kk);
        v2f b = *(const v2f*)(brow + kk);
        acc = __builtin_amdgcn_wmma_f32_16x16x4_f32(false, a, false, b, (short)0, acc, false, false);
    }
#pragma unroll
    for (int r = 0; r < 8; ++r) {
        size_t row = (size_t)(mt * 16 + hi * 8 + r);
        out[row * N_NODES + nt * 16 + lid] = acc[r];
    }
}

// ---------------- launcher ----------------

extern "C" void kernel_launch(void* const* d_in, const int* in_sizes, int n_in,
                              void* d_out, int out_size, void* d_ws, size_t ws_size,
                              hipStream_t stream) {
    const int*   x   = (const int*)d_in[0];
    const int*   ei  = (const int*)d_in[1];
    const float* ea  = (const float*)d_in[2];
    const float* emb = (const float*)d_in[3];
    const float* W1  = (const float*)d_in[4];
    const float* b1  = (const float*)d_in[5];
    const float* E1w = (const float*)d_in[6];
    const float* E1b = (const float*)d_in[7];
    const float* W2  = (const float*)d_in[8];
    const float* b2  = (const float*)d_in[9];
    const float* E2w = (const float*)d_in[10];
    const float* E2b = (const float*)d_in[11];
    float* out = (float*)d_out;

    float* ws   = (float*)d_ws;
    float* h0   = ws;                       // N*128
    float* h1   = h0 + N_NODES * EMBED;     // N*64
    float* h2   = h1 + N_NODES * HID1;      // N*32
    float* agg1 = h2 + N_NODES * HID2;      // N*128
    float* agg2 = agg1 + N_NODES * EMBED;   // N*64   (agg1+agg2 contiguous)

    int nzero = N_NODES * (EMBED + HID1);
    k_zero<<<(nzero + 255) / 256, 256, 0, stream>>>(agg1, nzero);
    k_gather<<<(N_NODES * EMBED) / 256, 256, 0, stream>>>(x, emb, h0);

    k_edge1<<<N_EDGES, EMBED, 0, stream>>>(ei, ea, E1w, E1b, h0, agg1);

    // 512*4 = 2048 waves, 8 waves/block -> 256 blocks (exact, EXEC all ones)
    k_update1<<<256, 256, 0, stream>>>(h0, agg1, W1, b1, h1);

    k_edge2<<<N_EDGES, HID1, 0, stream>>>(ei, ea, E2w, E2b, h1, agg2);

    // 512*2 = 1024 waves -> 128 blocks
    k_update2<<<128, 256, 0, stream>>>(h1, agg2, W2, b2, h2);

    // 512*512 = 262144 tiles -> 32768 blocks of 8 waves
    k_outer<<<32768, 256, 0, stream>>>(h2, out);
}